// TransformerLayer_81518479278129
// MI455X (gfx1250) — compile-verified
//
#include <hip/hip_runtime.h>
#include <hip/hip_bf16.h>
#include <math.h>

// ---------------------------------------------------------------------------
// CDNA5 (gfx1250) transformer layer, fp32 end-to-end via V_WMMA_F32_16X16X4_F32
// All five GEMM shapes reduced to a single xt-form kernel (both operands
// K-contiguous) by producing V transposed: Vt = Wv @ X^T (itself xt-form).
// ---------------------------------------------------------------------------

typedef float v2f __attribute__((ext_vector_type(2)));
typedef float v8f __attribute__((ext_vector_type(8)));

#define TT 8192
#define DD 1024

__device__ __forceinline__ v2f ld_f2(const float* p) {
    return *(const v2f*)p;   // 8B-aligned: even k offsets, 1024/8192 row strides
}

__device__ __forceinline__ v8f wmma_f32(v2f a, v2f b, v8f c) {
    // 8 args: (neg_a, A, neg_b, B, c_mod, C, reuse_a, reuse_b)
    return __builtin_amdgcn_wmma_f32_16x16x4_f32(false, a, false, b, (short)0, c, false, false);
}

__device__ __forceinline__ float gelu_exact(float x) {
    return 0.5f * x * (1.0f + erff(x * 0.70710678118654752f));
}

// ---------------------------------------------------------------------------
// GEMM "xt" form:  C[M,N] = A[M,KD] @ W[N,KD]^T      (lda = ldw = KD, ldc = N)
// EPI 0: C = acc * scale
// EPI 1: C = gelu(acc + bias[col])
// EPI 2: C = acc + bias[col] + res[row,col]
// EPI 3: C = acc + res[row,col]
// Block = 128 threads (4 waves, 2x2); block tile 128x128; wave tile 64x64
// (4x4 fragments of 16x16, K-step 4). KD compile-time -> immediate offsets.
// ---------------------------------------------------------------------------
template <int EPI, int KD>
__global__ __launch_bounds__(128)
void gemm_xt(const float* __restrict__ A, const float* __restrict__ W,
             const float* __restrict__ bias, const float* __restrict__ res,
             float* __restrict__ C, int N, float scale)
{
    const int lane = threadIdx.x & 31;
    const int wave = threadIdx.x >> 5;
    const int rowBase = blockIdx.y * 128 + (wave >> 1) * 64;
    const int colBase = blockIdx.x * 128 + (wave & 1) * 64;
    const int l15 = lane & 15;            // row-in-frag (A) / col-in-frag (B)
    const int kk  = (lane >> 4) << 1;     // lanes 0-15 -> K{0,1}; 16-31 -> K{2,3}

    v8f acc[4][4];
    #pragma unroll
    for (int mi = 0; mi < 4; ++mi)
        #pragma unroll
        for (int ni = 0; ni < 4; ++ni)
            acc[mi][ni] = (v8f){};

    const float* aPtr = A + (size_t)(rowBase + l15) * KD + kk;
    const float* bPtr = W + (size_t)(colBase + l15) * KD + kk;

    #pragma unroll 2
    for (int k0 = 0; k0 < KD; k0 += 4) {
        v2f a[4], b[4];
        #pragma unroll
        for (int i = 0; i < 4; ++i)
            a[i] = ld_f2(aPtr + k0 + (size_t)i * (16 * KD));   // compile-time strides
        #pragma unroll
        for (int i = 0; i < 4; ++i)
            b[i] = ld_f2(bPtr + k0 + (size_t)i * (16 * KD));
        #pragma unroll
        for (int mi = 0; mi < 4; ++mi)
            #pragma unroll
            for (int ni = 0; ni < 4; ++ni)
                acc[mi][ni] = wmma_f32(a[mi], b[ni], acc[mi][ni]);
    }

    const int rOff = (lane < 16) ? 0 : 8;
    #pragma unroll
    for (int mi = 0; mi < 4; ++mi) {
        #pragma unroll
        for (int ni = 0; ni < 4; ++ni) {
            #pragma unroll
            for (int g = 0; g < 8; ++g) {
                int r = rowBase + mi * 16 + g + rOff;
                int c = colBase + ni * 16 + l15;
                float v = acc[mi][ni][g];
                if (EPI == 0) {
                    v *= scale;
                } else if (EPI == 1) {
                    v = gelu_exact(v + bias[c]);
                } else if (EPI == 2) {
                    v = v + bias[c] + res[(size_t)r * N + c];
                } else {
                    v = v + res[(size_t)r * N + c];
                }
                C[(size_t)r * N + c] = v;
            }
        }
    }
}

// ---------------------------------------------------------------------------
// Row softmax over 8192 columns, one 256-thread block (8 waves) per row.
// ---------------------------------------------------------------------------
__global__ __launch_bounds__(256)
void softmax_rows(float* __restrict__ attn, int T)
{
    __shared__ float red[256];
    const int tid = threadIdx.x;
    float* p = attn + (size_t)blockIdx.x * T;

    float mx = -INFINITY;
    for (int i = tid; i < T; i += 256) mx = fmaxf(mx, p[i]);
    red[tid] = mx;
    __syncthreads();
    for (int s = 128; s > 0; s >>= 1) {
        if (tid < s) red[tid] = fmaxf(red[tid], red[tid + s]);
        __syncthreads();
    }
    mx = red[0];
    __syncthreads();

    float sum = 0.0f;
    for (int i = tid; i < T; i += 256) {
        float e = expf(p[i] - mx);
        p[i] = e;
        sum += e;
    }
    red[tid] = sum;
    __syncthreads();
    for (int s = 128; s > 0; s >>= 1) {
        if (tid < s) red[tid] += red[tid + s];
        __syncthreads();
    }
    const float inv = 1.0f / red[0];
    for (int i = tid; i < T; i += 256) p[i] *= inv;
}

// ---------------------------------------------------------------------------
// Host-side orchestration.
// d_in order: input_vecs, Wk, Wq, Wv, W1, b1, W2, b2  (all float32)
// d_out: out [T,D] then attn [T,T], concatenated flat.
// ---------------------------------------------------------------------------
extern "C" void kernel_launch(void* const* d_in, const int* in_sizes, int n_in,
                              void* d_out, int out_size, void* d_ws, size_t ws_size,
                              hipStream_t stream)
{
    (void)in_sizes; (void)n_in; (void)out_size; (void)ws_size;

    const float* X  = (const float*)d_in[0];
    const float* Wk = (const float*)d_in[1];
    const float* Wq = (const float*)d_in[2];
    const float* Wv = (const float*)d_in[3];
    const float* W1 = (const float*)d_in[4];
    const float* b1 = (const float*)d_in[5];
    const float* W2 = (const float*)d_in[6];
    const float* b2 = (const float*)d_in[7];

    const size_t TD = (size_t)TT * DD;
    float* ws  = (float*)d_ws;
    float* Qb  = ws;            // [T, D]
    float* Kb  = ws + TD;       // [T, D]
    float* Vt  = ws + 2 * TD;   // [D, T]  (V transposed)
    float* CTX = ws + 3 * TD;   // [T, D]
    float* Hb  = ws + 4 * TD;   // [T, D]

    float* outp = (float*)d_out;        // [T, D]
    float* attn = outp + TD;            // [T, T]

    dim3 blk(128);
    dim3 gTD(DD / 128, TT / 128);       // (8, 64)   for [T,D] outputs
    dim3 gDT(TT / 128, DD / 128);       // (64, 8)   for Vt [D,T]
    dim3 gTT(TT / 128, TT / 128);       // (64, 64)  for scores [T,T]

    // K = X @ Wk^T ; Q = X @ Wq^T
    gemm_xt<0, DD><<<gTD, blk, 0, stream>>>(X, Wk, nullptr, nullptr, Kb, DD, 1.0f);
    gemm_xt<0, DD><<<gTD, blk, 0, stream>>>(X, Wq, nullptr, nullptr, Qb, DD, 1.0f);
    // Vt = Wv @ X^T  (so that ctx GEMM is also xt-form with coalesced loads)
    gemm_xt<0, DD><<<gDT, blk, 0, stream>>>(Wv, X, nullptr, nullptr, Vt, TT, 1.0f);

    // scores = Q @ K^T / sqrt(D)
    gemm_xt<0, DD><<<gTT, blk, 0, stream>>>(Qb, Kb, nullptr, nullptr, attn, TT, 0.03125f);

    // softmax over keys (dim=1), in place -> attn output
    softmax_rows<<<dim3(TT), dim3(256), 0, stream>>>(attn, TT);

    // ctx = attn @ V + X  ==  attn[T,T] @ Vt[D,T]^T + X
    gemm_xt<3, TT><<<gTD, blk, 0, stream>>>(attn, Vt, nullptr, X, CTX, DD, 1.0f);

    // h = gelu(ctx @ W1^T + b1)   (exact erf GELU)
    gemm_xt<1, DD><<<gTD, blk, 0, stream>>>(CTX, W1, b1, nullptr, Hb, DD, 1.0f);

    // out = h @ W2^T + b2 + ctx
    gemm_xt<2, DD><<<gTD, blk, 0, stream>>>(Hb, W2, b2, CTX, outp, DD, 1.0f);
}